// BayesianChangePointDetector_50096498541205
// MI455X (gfx1250) — compile-verified
//
#include <hip/hip_runtime.h>

#define T_LEN 8192
#define N_DIM 32
#define MS 16
#define BLOCK 1024
#define NWAVE (BLOCK / 32)          // 32 waves
#define CHUNK (T_LEN / BLOCK)       // 8 elements per thread in the scan
#define NEAR_TH 6553                // int(8192 * 0.8)

typedef __attribute__((ext_vector_type(2))) float v2f;
typedef __attribute__((ext_vector_type(8))) float v8f;

__global__ __launch_bounds__(BLOCK) void bcpd_kernel(
    const float* __restrict__ x,
    const float* __restrict__ p_pm,
    const float* __restrict__ p_pv,
    const float* __restrict__ p_nv,
    float* __restrict__ out)
{
    // LDS: inclusive prefix sums of per-t mean and mean^2 (64 KB) + tiny scratch.
    __shared__ float s1[T_LEN];
    __shared__ float s2[T_LEN];
    __shared__ float wsum1[NWAVE];
    __shared__ float wsum2[NWAVE];
    __shared__ float wredA[NWAVE];
    __shared__ float wredB[NWAVE];

    const int b    = blockIdx.x;
    const int tid  = threadIdx.x;
    const int lane = tid & 31;
    const int wave = tid >> 5;
    const int m16  = lane & 15;   // M row within 16x16 tile
    const int half = lane >> 4;   // selects K-pair {0,1} vs {2,3} for A

    const float* xrow = x + (size_t)b * (T_LEN * N_DIM);

    // ---- Phase 1: mean over N=32 via chained V_WMMA_F32_16X16X4_F32 ----
    // A = x[16 t's x 4 n's] (f32), B = ones(4x16)  =>  D[i][*] = sum_n x[t_i, n]
    v2f onesB; onesB.x = 1.0f; onesB.y = 1.0f;
    for (int tile = wave; tile < T_LEN / 16; tile += NWAVE) {
        const int t0 = tile * 16;
        const float* base = xrow + (size_t)(t0 + m16) * N_DIM + 2 * half;
        // prefetch this wave's next tile (NWAVE tiles ahead => +16*NWAVE rows)
        __builtin_prefetch(base + 16 * NWAVE * N_DIM, 0, 0);
        v8f acc = {};
#pragma unroll
        for (int kk = 0; kk < 8; ++kk) {
            v2f a = *(const v2f*)(base + kk * 4);
            acc = __builtin_amdgcn_wmma_f32_16x16x4_f32(
                false, a, false, onesB, (short)0, acc, false, false);
        }
        // D layout: VGPR r, lanes 0-15 -> M=r ; lanes 16-31 -> M=r+8. All 16
        // columns identical; lanes with m16==0 own each row exactly once.
        if (m16 == 0) {
#pragma unroll
            for (int r = 0; r < 8; ++r)
                s1[t0 + 8 * half + r] = acc[r] * (1.0f / 32.0f);
        }
    }
    __syncthreads();

    // ---- Phase 2: hierarchical block-wide inclusive scans of m and m^2 ----
    const int cs = tid * CHUNK;
    float c1 = 0.0f, c2 = 0.0f;
#pragma unroll
    for (int i = 0; i < CHUNK; ++i) {
        float mv = s1[cs + i];
        c1 += mv;
        c2 += mv * mv;
        s1[cs + i] = c1;
        s2[cs + i] = c2;
    }
    // wave32 inclusive scan of per-thread chunk totals
    float lt1 = c1, lt2 = c2;
#pragma unroll
    for (int off = 1; off < 32; off <<= 1) {
        float u1 = __shfl_up(lt1, off, 32);
        float u2 = __shfl_up(lt2, off, 32);
        if (lane >= off) { lt1 += u1; lt2 += u2; }
    }
    if (lane == 31) { wsum1[wave] = lt1; wsum2[wave] = lt2; }
    __syncthreads();
    if (wave == 0) {  // wave 0 scans the 32 wave totals
        float w1 = wsum1[lane], w2 = wsum2[lane];
#pragma unroll
        for (int off = 1; off < 32; off <<= 1) {
            float u1 = __shfl_up(w1, off, 32);
            float u2 = __shfl_up(w2, off, 32);
            if (lane >= off) { w1 += u1; w2 += u2; }
        }
        wsum1[lane] = w1;
        wsum2[lane] = w2;
    }
    __syncthreads();
    {
        float o1 = (lt1 - c1) + (wave > 0 ? wsum1[wave - 1] : 0.0f);
        float o2 = (lt2 - c2) + (wave > 0 ? wsum2[wave - 1] : 0.0f);
#pragma unroll
        for (int i = 0; i < CHUNK; ++i) {
            s1[cs + i] += o1;
            s2[cs + i] += o2;
        }
    }
    __syncthreads();

    // ---- Phase 3: Bayes factors, softmax, output ----
    const float pm = p_pm[0];
    const float pv = log1pf(expf(p_pv[0]));   // softplus
    const float nv = log1pf(expf(p_nv[0]));   // softplus
    const float inv_pv = 1.0f / pv;
    const float inv_nv = 1.0f / nv;
    const float tpnl = logf(6.283185307179586f * nv);  // log(2*pi*nv)

    const float totS = s1[T_LEN - 1];
    const float totQ = s2[T_LEN - 1];

    auto log_ml = [&](float n, float mean, float var) -> float {
        float post_var  = 1.0f / (inv_pv + n * inv_nv);
        float post_mean = post_var * (pm * inv_pv + n * mean * inv_nv);
        return -0.5f * n * tpnl + 0.5f * logf(post_var * inv_pv)
             - 0.5f * (n * var * inv_nv + mean * mean * n * inv_nv
                       - post_mean * post_mean / post_var + pm * pm * inv_pv);
    };

    const float meanW = totS / (float)T_LEN;
    const float varW  = fmaxf((totQ - totS * totS / (float)T_LEN) / (float)(T_LEN - 1), 1e-8f);
    const float lmlW  = log_ml((float)T_LEN, meanW, varW);

    auto bf_at = [&](int p) -> float {
        float nL = (float)p, nR = (float)(T_LEN - p);
        float sumL = s1[p - 1], sqL = s2[p - 1];
        float sumR = totS - sumL, sqR = totQ - sqL;
        float mL = sumL / nL, mR = sumR / nR;
        float vL = fmaxf((sqL - sumL * sumL / nL) / (nL - 1.0f), 1e-8f);
        float vR = fmaxf((sqR - sumR * sumR / nR) / (nR - 1.0f), 1e-8f);
        return log_ml(nL, mL, vL) + log_ml(nR, mR, vR) - lmlW;
    };

    // Pass A: global max of bf
    float bfmax = -3.4e38f;
    for (int p = MS + tid; p < T_LEN - MS; p += BLOCK)
        bfmax = fmaxf(bfmax, bf_at(p));
#pragma unroll
    for (int off = 16; off >= 1; off >>= 1)
        bfmax = fmaxf(bfmax, __shfl_xor(bfmax, off, 32));
    if (lane == 0) wredA[wave] = bfmax;
    __syncthreads();
    {
        float mx = wredA[0];
#pragma unroll
        for (int w = 1; w < NWAVE; ++w) mx = fmaxf(mx, wredA[w]);
        bfmax = mx;
    }
    __syncthreads();

    // Pass B: softmax denominator + near-end mass (bf recomputed; ALU is free here)
    float tot = 0.0f, nearw = 0.0f;
    for (int p = MS + tid; p < T_LEN - MS; p += BLOCK) {
        float e = expf(bf_at(p) - bfmax);
        tot += e;
        if (p > NEAR_TH) nearw += e;
    }
#pragma unroll
    for (int off = 16; off >= 1; off >>= 1) {
        tot   += __shfl_xor(tot, off, 32);
        nearw += __shfl_xor(nearw, off, 32);
    }
    if (lane == 0) { wredA[wave] = tot; wredB[wave] = nearw; }
    __syncthreads();
    if (tid == 0) {
        float st = 0.0f, sn = 0.0f;
#pragma unroll
        for (int w = 0; w < NWAVE; ++w) { st += wredA[w]; sn += wredB[w]; }
        float conf = 1.0f / (1.0f + expf(-bfmax));
        out[b] = conf * (sn / st);
    }
}

extern "C" void kernel_launch(void* const* d_in, const int* in_sizes, int n_in,
                              void* d_out, int out_size, void* d_ws, size_t ws_size,
                              hipStream_t stream) {
    const float* x    = (const float*)d_in[0];
    const float* ppm  = (const float*)d_in[1];
    const float* ppv  = (const float*)d_in[2];
    const float* pnv  = (const float*)d_in[3];
    float* out = (float*)d_out;
    const int B = out_size;  // 128 rows, one workgroup each
    bcpd_kernel<<<B, BLOCK, 0, stream>>>(x, ppm, ppv, pnv, out);
}